// avg_remover_34514357191115
// MI455X (gfx1250) — compile-verified
//
#include <hip/hip_runtime.h>
#include <stdint.h>

typedef float v2f __attribute__((ext_vector_type(2)));
typedef float v8f __attribute__((ext_vector_type(8)));

#define B_ROWS   8192
#define D_COLS   8192
#define N_EXP    10
#define TILE_M   32          // 2 WMMA M-tiles per block
#define CHUNK_K  128         // K per staged LDS chunk
#define STRIDE   132         // CHUNK_K+4 pad dwords; %64==4 -> conflict-free ds_load_b64
#define BLOCK    256         // 8 waves: waves 0-3 -> tile 0, waves 4-7 -> tile 1
#define NCHUNK   (D_COLS / CHUNK_K)   // 64

// GLOBAL_LOAD_ASYNC_TO_LDS_B128: per-lane 16B global->LDS copy, tracked by ASYNCcnt.
__device__ __forceinline__ void async_b128(uint32_t lds_addr, const float* g) {
    asm volatile("global_load_async_to_lds_b128 %0, %1, off"
                 :: "v"(lds_addr), "v"((uint64_t)(uintptr_t)g)
                 : "memory");
}
__device__ __forceinline__ void wait_async0() {
#if __has_builtin(__builtin_amdgcn_s_wait_asynccnt)
    __builtin_amdgcn_s_wait_asynccnt(0);
#else
    asm volatile("s_wait_asynccnt 0x0" ::: "memory");
#endif
}

__global__ __launch_bounds__(BLOCK)
void expert_dot_wmma_async(const float* __restrict__ X,
                           const int*   __restrict__ label,
                           const float* __restrict__ W,
                           float*       __restrict__ out)
{
    __shared__ float lds_x[2][TILE_M * STRIDE];   // double-buffered X tile (32 x 128)
    __shared__ float lds_w[2][16 * STRIDE];       // double-buffered W tile (16 x 128)
    __shared__ float red[2][4][256];              // per-tile per-wave partial 16x16
    __shared__ float tileout[2][256];             // reduced 16x16 tiles

    const int t       = threadIdx.x;
    const int wave    = t >> 5;
    const int lane    = t & 31;
    const int tid16   = lane & 15;
    const int kofs    = (lane >> 4) << 1;   // 0 or 2 (K half-wave split)
    const int tile_id = wave >> 2;          // which 16-row M-tile
    const int wsub    = wave & 3;           // K-partition within the tile
    const int row0    = blockIdx.x * TILE_M;

    // ---- async staging: fully uniform (padded W rows clamp to a real row;
    //      columns 10..15 of the result are never selected since label < 10)
    auto stage = [&](int buf, int k0) {
        #pragma unroll
        for (int i = 0; i < 4; ++i) {                 // X: 32 rows x 32 float4
            int idx = i * BLOCK + t;
            int r = idx >> 5, c4 = idx & 31;
            async_b128((uint32_t)(uintptr_t)&lds_x[buf][r * STRIDE + c4 * 4],
                       &X[(size_t)(row0 + r) * D_COLS + k0 + c4 * 4]);
        }
        #pragma unroll
        for (int i = 0; i < 2; ++i) {                 // W: 16 rows x 32 float4
            int idx = i * BLOCK + t;
            int r = idx >> 5, c4 = idx & 31;
            int wr = (r < N_EXP) ? r : (N_EXP - 1);
            async_b128((uint32_t)(uintptr_t)&lds_w[buf][r * STRIDE + c4 * 4],
                       &W[(size_t)wr * D_COLS + k0 + c4 * 4]);
        }
    };

    v8f acc = {};   // 16x16 f32 C/D tile (8 VGPRs/lane)

    stage(0, 0);                // prologue: fill buffer 0
    wait_async0();
    __syncthreads();

    for (int c = 0; c < NCHUNK; ++c) {
        const int cur = c & 1;
        if (c + 1 < NCHUNK)
            stage(cur ^ 1, (c + 1) * CHUNK_K);   // prefetch next chunk async

        // A 16x4 f32: lanes 0-15 -> M=lane, K=k..k+1; lanes 16-31 -> M=lane-16, K=k+2..k+3
        // B 4x16 f32 mirrors it per column; both read 8B/lane from padded LDS rows.
        const float* xa = &lds_x[cur][(tile_id * 16 + tid16) * STRIDE];
        const float* wb = &lds_w[cur][tid16 * STRIDE];
        #pragma unroll
        for (int s = 0; s < 8; ++s) {
            int kk = wsub * 32 + s * 4 + kofs;
            v2f a = *(const v2f*)&xa[kk];
            v2f b = *(const v2f*)&wb[kk];
            acc = __builtin_amdgcn_wmma_f32_16x16x4_f32(
                      /*neg_a=*/false, a, /*neg_b=*/false, b,
                      /*c_mod=*/(short)0, acc,
                      /*reuse_a=*/false, /*reuse_b=*/false);
        }

        wait_async0();      // next-chunk copies landed (no-op on last iter)
        __syncthreads();    // all waves done reading `cur` + all copies visible
    }

    // ---- deterministic cross-wave reduction (4 waves per tile)
    #pragma unroll
    for (int r = 0; r < 8; ++r)
        red[tile_id][wsub][lane * 8 + r] = acc[r];
    __syncthreads();

    {
        int l = t >> 3, rr = t & 7;     // (lane, vgpr) slot
        #pragma unroll
        for (int tt = 0; tt < 2; ++tt) {
            float s = 0.f;
            #pragma unroll
            for (int w = 0; w < 4; ++w)
                s += red[tt][w][l * 8 + rr];
            // C/D layout: lanes 0-15 -> (M=rr, N=l); lanes 16-31 -> (M=rr+8, N=l-16)
            int m = (l < 16) ? rr : rr + 8;
            int n = (l < 16) ? l  : l - 16;
            tileout[tt][m * 16 + n] = s;
        }
    }
    __syncthreads();

    // ---- gather-select the labeled expert column
    if (t < TILE_M) {
        int lbl = label[row0 + t];                       // 0..9
        out[row0 + t] = tileout[t >> 4][(t & 15) * 16 + lbl];
    }
}

extern "C" void kernel_launch(void* const* d_in, const int* in_sizes, int n_in,
                              void* d_out, int out_size, void* d_ws, size_t ws_size,
                              hipStream_t stream)
{
    (void)in_sizes; (void)n_in; (void)out_size; (void)d_ws; (void)ws_size;
    const float* X     = (const float*)d_in[0];
    const int*   label = (const int*)d_in[1];
    const float* W     = (const float*)d_in[2];
    float*       out   = (float*)d_out;

    expert_dot_wmma_async<<<dim3(B_ROWS / TILE_M), dim3(BLOCK), 0, stream>>>(X, label, W, out);
}